// Anchor_dp_18279380812451
// MI455X (gfx1250) — compile-verified
//
#include <hip/hip_runtime.h>
#include <hip/hip_bf16.h>
#include <math.h>

typedef __attribute__((ext_vector_type(2))) float v2f;
typedef __attribute__((ext_vector_type(8))) float v8f;

#define L_SEQ   262144
#define KLEN    64
#define TILE    256      // output columns per tile
#define HALO    256      // influence radius bound: |F|<=64, gap=1 -> 256 cols
#define WIN     512      // TILE + HALO columns processed per tile
#define CPL     16       // columns per lane (32 lanes * 16 = 512)
#define NBIG    (-1.0e30f)
#define NTILES  (L_SEQ / TILE)   // 1024
#define NMOMB   512              // blocks for moments pass

// ---------------------------------------------------------------------------
// Kernel 1: per-tile DP.  One wave32 per tile.  S = p^T * seq computed with
// v_wmma_f32_16x16x4_f32 (exact for one-hot seq), staged per 16-row strip in
// LDS; DP rows consume it with register-resident state + shuffle scans.
// ---------------------------------------------------------------------------
__global__ __launch_bounds__(32) void dp_kernel(
    const float* __restrict__ patterns,   // [16,4,64]
    const float* __restrict__ bias,       // [16]
    const float* __restrict__ data,       // [1,4,L]
    const int*   __restrict__ anchor,     // [1]
    float* __restrict__ out,              // d_out: [skew,kurt, ch(L)]
    float* __restrict__ tileSum,          // [NTILES]
    float* __restrict__ tileMax)          // [NTILES]
{
    __shared__ float pn[4 * KLEN];        // normalized pattern
    __shared__ int   baseS[WIN];          // base index per window column (-1 = phantom)
    __shared__ float Sst[16 * WIN];       // S strip: 16 pattern rows x WIN cols (32KB)

    const int lane = threadIdx.x;
    const int tile = blockIdx.x;
    const int ws   = tile * TILE - HALO;  // window start (global col), may be negative
    const int a    = anchor[0];
    const float bval = bias[a];

    // Normalize pattern columns (L2 over the 4 bases), NORM_RATIO = 1
    for (int i = lane; i < KLEN; i += 32) {
        float p0 = patterns[a * 256 + 0 * 64 + i];
        float p1 = patterns[a * 256 + 1 * 64 + i];
        float p2 = patterns[a * 256 + 2 * 64 + i];
        float p3 = patterns[a * 256 + 3 * 64 + i];
        float nrm = fmaxf(sqrtf(p0 * p0 + p1 * p1 + p2 * p2 + p3 * p3), 1e-12f);
        float inv = 1.0f / nrm;
        pn[0 * 64 + i] = p0 * inv;
        pn[1 * 64 + i] = p1 * inv;
        pn[2 * 64 + i] = p2 * inv;
        pn[3 * 64 + i] = p3 * inv;
    }
    // Decode one-hot sequence to base indices for this window
    for (int c = 0; c < CPL; ++c) {
        int wcol = lane * CPL + c;
        int g = ws + wcol;
        int b = -1;
        if (g >= 0) {
            float d1 = data[1 * L_SEQ + g];
            float d2 = data[2 * L_SEQ + g];
            float d3 = data[3 * L_SEQ + g];
            b = (d1 > 0.5f ? 1 : 0) + (d2 > 0.5f ? 2 : 0) + (d3 > 0.5f ? 3 : 0);
        }
        baseS[wcol] = b;
    }
    __syncthreads();

    const int lo = lane & 15;
    const int hi = lane >> 4;

    // G[0, c] = 0 everywhere (free start in sequence)
    float prevG[CPL];
#pragma unroll
    for (int c = 0; c < CPL; ++c) prevG[c] = 0.0f;

    for (int strip = 0; strip < 4; ++strip) {
        // A-matrix regs (16x4 f32): lane<16 holds {K0,K1}, lane>=16 holds {K2,K3}
        int row = strip * 16 + lo;
        v2f av;
        av.x = pn[(2 * hi + 0) * 64 + row];
        av.y = pn[(2 * hi + 1) * 64 + row];

        __syncthreads();   // previous strip's readers done before overwriting Sst
        for (int g2 = 0; g2 < WIN / 16; ++g2) {
            int b = baseS[g2 * 16 + lo];
            // B-matrix (4x16 f32) mirrored layout: lane<16 {K0,K1}, lane>=16 {K2,K3}
            v2f bv;
            bv.x = (b == 2 * hi + 0) ? 1.0f : 0.0f;
            bv.y = (b == 2 * hi + 1) ? 1.0f : 0.0f;
            v8f cz = {};
            v8f cr = __builtin_amdgcn_wmma_f32_16x16x4_f32(
                false, av, false, bv, (short)0, cz, false, false);
#pragma unroll
            for (int v = 0; v < 8; ++v)
                Sst[(v + 8 * hi) * WIN + g2 * 16 + lo] = cr[v];
        }
        __syncthreads();

        // 16 DP rows of this strip; DP row i = strip*16 + li + 1 uses S row li
        for (int li = 0; li < 16; ++li) {
            float leftPrev = __shfl_up(prevG[CPL - 1], 1);
            if (lane == 0) leftPrev = NBIG;   // unknown left boundary (halo absorbs)

            float f[CPL];
            float carry = NBIG;
#pragma unroll
            for (int c = 0; c < CPL; ++c) {
                int wcol = lane * CPL + c;
                float s = Sst[li * WIN + wcol];
                if (ws + wcol < 0) s = NBIG;  // phantom cols: no match allowed
                float diag = ((c == 0) ? leftPrev : prevG[c - 1]) + s;
                float A = fmaxf(diag, prevG[c] - 1.0f);
                carry = fmaxf(A, carry - 1.0f);   // local max-plus scan
                f[c] = carry;
            }
            // cross-lane max-plus scan of chunk-end values (decay 16/lane)
            float e = f[CPL - 1];
#pragma unroll
            for (int d = 1; d < 32; d <<= 1) {
                float o = __shfl_up(e, (unsigned)d);
                o = (lane >= d) ? o : NBIG;
                e = fmaxf(e, o - (float)(d * CPL));
            }
            float cin = __shfl_up(e, 1);
            if (lane == 0) cin = NBIG;
#pragma unroll
            for (int c = 0; c < CPL; ++c)
                prevG[c] = fmaxf(f[c], cin - (float)(c + 1));
        }
    }

    // Epilogue: ch = leakyrelu(asm + bias); lanes 16..31 own the output tile
    float lsum = 0.0f, lmax = -3.4e38f;
#pragma unroll
    for (int c = 0; c < CPL; ++c) {
        float ch = prevG[c] + bval;
        ch = (ch >= 0.0f) ? ch : 0.001f * ch;
        if (hi) {
            int gcol = ws + lane * CPL + c;   // in [tile*TILE, tile*TILE+TILE)
            out[2 + gcol] = ch;
            lsum += fabsf(ch);
            lmax = fmaxf(lmax, ch);
        }
    }
#pragma unroll
    for (int d = 16; d >= 1; d >>= 1) {
        lsum += __shfl_xor(lsum, d);
        lmax = fmaxf(lmax, __shfl_xor(lmax, d));
    }
    if (lane == 0) { tileSum[tile] = lsum; tileMax[tile] = lmax; }
}

// ---------------------------------------------------------------------------
// Kernel 2: combine per-tile partials -> l1 (clamped at eps) and global max.
// ---------------------------------------------------------------------------
__global__ __launch_bounds__(256) void reduce_tiles(
    const float* __restrict__ tileSum, const float* __restrict__ tileMax,
    float* __restrict__ l1max)
{
    __shared__ float ssum[256];
    __shared__ float smax[256];
    int t = threadIdx.x;
    float s = 0.0f, m = -3.4e38f;
    for (int i = t; i < NTILES; i += 256) {
        s += tileSum[i];
        m = fmaxf(m, tileMax[i]);
    }
    ssum[t] = s; smax[t] = m;
    __syncthreads();
    for (int off = 128; off; off >>= 1) {
        if (t < off) {
            ssum[t] += ssum[t + off];
            smax[t] = fmaxf(smax[t], smax[t + off]);
        }
        __syncthreads();
    }
    if (t == 0) {
        l1max[0] = fmaxf(ssum[0], 1e-12f);  // l1 norm (reference eps clamp)
        l1max[1] = smax[0];                  // max(ch) for softmax shift
    }
}

// ---------------------------------------------------------------------------
// Kernel 3: softmax moments in double: Z = sum e, Sx = sum e*j, Sxx = sum e*j^2
// with e = exp((ch - chmax)/l1).
// ---------------------------------------------------------------------------
__global__ __launch_bounds__(256) void moments_kernel(
    const float* __restrict__ ch, const float* __restrict__ l1max,
    double* __restrict__ part)
{
    __shared__ double sZ[256], sX[256], sXX[256];
    int t = threadIdx.x;
    double inv = 1.0 / (double)l1max[0];
    double mx  = (double)l1max[1];
    double Z = 0.0, Sx = 0.0, Sxx = 0.0;
    for (int j = blockIdx.x * 256 + t; j < L_SEQ; j += gridDim.x * 256) {
        double e = exp(((double)ch[j] - mx) * inv);
        double x = (double)j;
        Z += e; Sx += e * x; Sxx += e * x * x;
    }
    sZ[t] = Z; sX[t] = Sx; sXX[t] = Sxx;
    __syncthreads();
    for (int off = 128; off; off >>= 1) {
        if (t < off) {
            sZ[t] += sZ[t + off];
            sX[t] += sX[t + off];
            sXX[t] += sXX[t + off];
        }
        __syncthreads();
    }
    if (t == 0) {
        part[blockIdx.x]             = sZ[0];
        part[NMOMB + blockIdx.x]     = sX[0];
        part[2 * NMOMB + blockIdx.x] = sXX[0];
    }
}

// ---------------------------------------------------------------------------
// Kernel 4: final combine; mean/var from softmax moments, skew/kurt via
// closed-form power sums sum j^p (z-moments are plain means over positions).
// ---------------------------------------------------------------------------
__global__ __launch_bounds__(256) void finalize_kernel(
    const double* __restrict__ part, float* __restrict__ out)
{
    __shared__ double sZ[256], sX[256], sXX[256];
    int t = threadIdx.x;
    double Z = 0.0, Sx = 0.0, Sxx = 0.0;
    for (int i = t; i < NMOMB; i += 256) {
        Z   += part[i];
        Sx  += part[NMOMB + i];
        Sxx += part[2 * NMOMB + i];
    }
    sZ[t] = Z; sX[t] = Sx; sXX[t] = Sxx;
    __syncthreads();
    for (int off = 128; off; off >>= 1) {
        if (t < off) {
            sZ[t] += sZ[t + off];
            sX[t] += sX[t + off];
            sXX[t] += sXX[t + off];
        }
        __syncthreads();
    }
    if (t == 0) {
        double Ld  = (double)L_SEQ;
        double m   = sX[0] / sZ[0];
        double var = sXX[0] / sZ[0] - m * m;
        double sig = sqrt(var);
        double S1 = Ld * (Ld - 1.0) * 0.5;
        double S2 = Ld * (Ld - 1.0) * (2.0 * Ld - 1.0) / 6.0;
        double S3 = S1 * S1;
        double S4 = (Ld - 1.0) * Ld * (2.0 * Ld - 1.0)
                    * (3.0 * Ld * Ld - 3.0 * Ld - 1.0) / 30.0;
        double m2 = m * m, m3 = m2 * m, m4 = m3 * m;
        double C3 = S3 - 3.0 * m * S2 + 3.0 * m2 * S1 - Ld * m3;
        double C4 = S4 - 4.0 * m * S3 + 6.0 * m2 * S2 - 4.0 * m3 * S1 + Ld * m4;
        double s3 = sig * sig * sig;
        out[0] = (float)(C3 / (Ld * s3));          // skew = mean(z^3)
        out[1] = (float)(C4 / (Ld * s3 * sig) - 3.0); // kurt = mean(z^4) - 3
    }
}

// ---------------------------------------------------------------------------
extern "C" void kernel_launch(void* const* d_in, const int* in_sizes, int n_in,
                              void* d_out, int out_size, void* d_ws, size_t ws_size,
                              hipStream_t stream)
{
    (void)in_sizes; (void)n_in; (void)out_size; (void)ws_size;
    const float* patterns = (const float*)d_in[0];
    const float* bias     = (const float*)d_in[1];
    const float* data     = (const float*)d_in[2];
    const int*   anchor   = (const int*)d_in[3];
    float* out = (float*)d_out;

    char* ws = (char*)d_ws;
    float*  tileSum = (float*)(ws + 0);          // NTILES floats
    float*  tileMax = (float*)(ws + 4096);       // NTILES floats
    float*  l1max   = (float*)(ws + 8192);       // 2 floats
    double* parts   = (double*)(ws + 8704);      // 3*NMOMB doubles (8-aligned)

    dp_kernel<<<NTILES, 32, 0, stream>>>(patterns, bias, data, anchor,
                                         out, tileSum, tileMax);
    reduce_tiles<<<1, 256, 0, stream>>>(tileSum, tileMax, l1max);
    moments_kernel<<<NMOMB, 256, 0, stream>>>(out + 2, l1max, parts);
    finalize_kernel<<<1, 256, 0, stream>>>(parts, out);
}